// WSDEncoder2_62577673502770
// MI455X (gfx1250) — compile-verified
//
#include <hip/hip_runtime.h>
#include <hip/hip_bf16.h>

// Masked BPE-span average pooling on MI455X (gfx1250).
//   hidden : [B, S, H] f32     (d_in[0])
//   att    : [B, S]    bool    (d_in[1])  -- redundant, unused (derivable from span)
//   span   : [B, S, 2] i64     (d_in[2])
//   out    : [B, S-2, H] f32   (d_out)    out[b,t-1,:] = mean(hidden[b, first:last, :]) or 0
//
// Bandwidth-bound (~512 MB traffic -> ~22 us @ 23.3 TB/s). Data path uses the
// CDNA5 async pipe: GLOBAL_LOAD_ASYNC_TO_LDS_B128 (ASYNCcnt, th:NT) stages the
// 1..3 source rows per token into LDS, s_wait_asynccnt 0, then each thread sums
// its own 16B slices back out of LDS. Each wave reads only bytes written by its
// own async instructions, so no workgroup barrier is required.

typedef float v4f __attribute__((ext_vector_type(4)));

#define BB 32
#define SS 2048
#define HH 1024
#define SOUT (SS - 2)
#define THREADS 256          // 8 wave32; 256 float4 == one H row
#define MAX_SPAN 3           // span lengths are in {1,2,3} by construction

__global__ __launch_bounds__(THREADS) void
spanpool_kernel(const float* __restrict__ hidden,
                const long long* __restrict__ span,
                float* __restrict__ dst)
{
    __shared__ float lds[MAX_SPAN * HH];     // 12 KB staging: up to 3 rows

    const int tm1 = blockIdx.x;              // t - 1, in [0, SOUT)
    const int b   = blockIdx.y;              // batch row
    const int t   = tm1 + 1;                 // token slot in [1, S-1)
    const int tid = threadIdx.x;

    // Uniform per-block span fetch -> scalar s_load (address depends only on blockIdx).
    const long long* sp = span + (((size_t)b * SS + t) << 1);   // 16B-aligned pair
    const long long first = sp[0];
    const long long last  = sp[1];
    const int n = (int)(last - first);

    v4f acc = (v4f){0.f, 0.f, 0.f, 0.f};

    if (n > 0) {
        const float* gbase = hidden + ((size_t)b * SS + (size_t)first) * HH + (size_t)tid * 4;

        if (n <= MAX_SPAN) {
            // ---- CDNA5 async global->LDS staging (16 B per lane per row) ----
            // Streams have zero reuse -> non-temporal in both WGP$ and L2.
            #pragma unroll
            for (int r = 0; r < MAX_SPAN; ++r) {
                if (r < n) {
                    unsigned lofs = (unsigned)(size_t)(&lds[r * HH + tid * 4]); // raw LDS byte addr
                    const float* g = gbase + (size_t)r * HH;
                    asm volatile("global_load_async_to_lds_b128 %0, %1, off th:TH_LOAD_NT"
                                 :: "v"(lofs), "v"(g)
                                 : "memory");
                }
            }
            // Wait for this wave's async transfers (ASYNCcnt), then read back.
            asm volatile("s_wait_asynccnt 0" ::: "memory");
            #pragma unroll
            for (int r = 0; r < MAX_SPAN; ++r) {
                if (r < n) {
                    v4f v = *(const v4f*)(&lds[r * HH + tid * 4]);
                    acc += v;
                }
            }
        } else {
            // Defensive fallback (never taken for spec inputs): direct global sum.
            #pragma clang loop unroll(disable)
            for (int r = 0; r < n; ++r) {
                v4f v = *(const v4f*)(gbase + (size_t)r * HH);
                acc += v;
            }
        }

        // Match reference: true f32 division by span length.
        const float fn = (float)n;
        acc.x = acc.x / fn;
        acc.y = acc.y / fn;
        acc.z = acc.z / fn;
        acc.w = acc.w / fn;
    }
    // n == 0 (padded/empty span): acc stays zero; every output element gets written.

    v4f* o = (v4f*)(dst + ((size_t)b * SOUT + (size_t)tm1) * HH) + tid;
    __builtin_nontemporal_store(acc, o);     // streamed output, never re-read
}

extern "C" void kernel_launch(void* const* d_in, const int* in_sizes, int n_in,
                              void* d_out, int out_size, void* d_ws, size_t ws_size,
                              hipStream_t stream)
{
    const float*     hidden = (const float*)d_in[0];
    // d_in[1] (att_mask) intentionally unused: span fully determines the result.
    const long long* span   = (const long long*)d_in[2];
    float*           out    = (float*)d_out;

    dim3 grid(SOUT, BB);     // 2046 x 32 blocks: one token per block, no index division
    dim3 block(THREADS);
    spanpool_kernel<<<grid, block, 0, stream>>>(hidden, span, out);
}